// MultiPrototypeLayer_23897198035269
// MI455X (gfx1250) — compile-verified
//
#include <hip/hip_runtime.h>

typedef __attribute__((ext_vector_type(2))) float v2f;
typedef __attribute__((ext_vector_type(8))) float v8f;

#define K_DIM   512
#define C_CLS   1000
#define P_PER   5
#define B_ROWS  4096
#define R_FRAG  4                      // 16-row A-fragments per wave -> 64 rows
#define OUT_HALF (B_ROWS * C_CLS)

// ---------------------------------------------------------------------------
// Wave-per-row sum of squares: dst[r] = sum_k src[r*K_DIM+k]^2   (wave32)
// ---------------------------------------------------------------------------
__global__ void rowsq_kernel(const float* __restrict__ src,
                             float* __restrict__ dst, int rows) {
  int wave = (blockIdx.x * blockDim.x + threadIdx.x) >> 5;
  int lane = threadIdx.x & 31;
  if (wave >= rows) return;
  const float* r = src + (size_t)wave * K_DIM;
  float s = 0.f;
  for (int k = lane; k < K_DIM; k += 32) {
    float v = r[k];
    s = fmaf(v, v, s);
  }
  // wave32 butterfly reduction
  #pragma unroll
  for (int off = 16; off > 0; off >>= 1) s += __shfl_xor(s, off, 32);
  if (lane == 0) dst[wave] = s;
}

// ---------------------------------------------------------------------------
// Main kernel: each wave computes a 64(batch) x 16(class) output tile.
// Four A-fragments (row blocks m0+{0,16,32,48}) share every B-fragment:
// each K-step issues 9 loads for 20 V_WMMA_F32_16X16X4_F32 (0.45 loads/WMMA).
// Prototype p of class (c0+n) lives in acc[*][p], column n; the min over
// prototypes is an elementwise min across the 5 accumulators.
//
// A-frag layout (ISA 7.12.2, 32-bit A 16x4): lane L holds row M = L&15,
//   VGPR0/1 = K = 2*(L>>4) + {0,1}  -> contiguous float2 in K-major x.
// B-frag layout (4x16, symmetric):   lane L holds col N = L&15,
//   VGPR0/1 = K = 2*(L>>4) + {0,1}  -> contiguous float2 in K-major protos.
// ---------------------------------------------------------------------------
__global__ __launch_bounds__(128) void proto_dist_kernel(
    const float* __restrict__ x, const float* __restrict__ proto,
    const float* __restrict__ xsq, const float* __restrict__ psq,
    float* __restrict__ out) {
  const int lane = threadIdx.x & 31;
  const int wave = threadIdx.x >> 5;
  const int m0   = (blockIdx.x * 4 + wave) * (R_FRAG * 16); // 64-row tile base
  const int c0   = blockIdx.y * 16;                          // class-tile base

  const int half = lane >> 4;   // 0 or 1 (second half-wave)
  const int lid  = lane & 15;
  const int koff = half * 2;    // K offset carried by this half-wave

  // A operands: rows m0+lid, m0+16+lid, m0+32+lid, m0+48+lid of x.
  const float* abase = x + (size_t)(m0 + lid) * K_DIM + koff;

  // B operand: column lid maps to class c0+lid; clamp for the ragged last
  // class tile (EXEC must stay all-ones for WMMA; stores are masked later).
  const int c     = c0 + lid;
  const int cc    = (c < C_CLS) ? c : (C_CLS - 1);
  const float* bbase = proto + ((size_t)cc * P_PER) * K_DIM + koff;

  v8f acc[R_FRAG][P_PER];
  #pragma unroll
  for (int r = 0; r < R_FRAG; ++r)
    #pragma unroll
    for (int t = 0; t < P_PER; ++t)
      acc[r][t] = (v8f){0.f,0.f,0.f,0.f,0.f,0.f,0.f,0.f};

  // 128 K-steps of 4; 4 A loads + 5 B loads feed 20 WMMAs per step.
  #pragma unroll 2
  for (int k = 0; k < K_DIM; k += 4) {
    v2f a[R_FRAG];
    #pragma unroll
    for (int r = 0; r < R_FRAG; ++r)
      a[r] = *(const v2f*)(abase + (size_t)r * 16 * K_DIM + k);
    #pragma unroll
    for (int t = 0; t < P_PER; ++t) {
      v2f b = *(const v2f*)(bbase + (size_t)t * K_DIM + k);
      #pragma unroll
      for (int r = 0; r < R_FRAG; ++r) {
        acc[r][t] = __builtin_amdgcn_wmma_f32_16x16x4_f32(
            false, a[r], false, b, (short)0, acc[r][t], false, false);
      }
    }
  }

  // ||p||^2 for the 5 prototypes of this lane's class (contiguous scalars).
  float ps[P_PER];
  #pragma unroll
  for (int t = 0; t < P_PER; ++t) ps[t] = psq[cc * P_PER + t];

  const bool valid = (c < C_CLS);

  // C/D layout (ISA 7.12.2): VGPR v, lane L -> M = v + 8*(L>>4), N = L&15.
  #pragma unroll
  for (int r = 0; r < R_FRAG; ++r) {
    #pragma unroll
    for (int v = 0; v < 8; ++v) {
      const int m = m0 + r * 16 + half * 8 + v;
      float dmin = 3.4e38f;
      #pragma unroll
      for (int t = 0; t < P_PER; ++t) {
        float d = fmaf(-2.0f, acc[r][t][v], ps[t]);   // -2*S + ||p||^2
        dmin = fminf(dmin, d);
      }
      dmin += xsq[m];                                  // + ||x||^2
      if (valid) {
        out[(size_t)m * C_CLS + c]            = -dmin;  // output 0: -min_dist
        out[(size_t)m * C_CLS + c + OUT_HALF] =  dmin;  // output 1:  min_dist
      }
    }
  }
}

// ---------------------------------------------------------------------------
extern "C" void kernel_launch(void* const* d_in, const int* in_sizes, int n_in,
                              void* d_out, int out_size, void* d_ws, size_t ws_size,
                              hipStream_t stream) {
  const float* x     = (const float*)d_in[0];   // [4096, 512]
  const float* proto = (const float*)d_in[1];   // [1000, 5, 512]
  float* out = (float*)d_out;                   // [2 * 4096 * 1000]
  float* ws  = (float*)d_ws;
  float* xsq = ws;                               // 4096 floats
  float* psq = ws + B_ROWS;                      // 5000 floats

  (void)in_sizes; (void)n_in; (void)out_size; (void)ws_size;

  // ||x_b||^2 : 4096 rows, one wave each
  rowsq_kernel<<<dim3((B_ROWS * 32 + 255) / 256), 256, 0, stream>>>(x, xsq, B_ROWS);
  // ||p_{c,p}||^2 : 5000 rows
  rowsq_kernel<<<dim3((C_CLS * P_PER * 32 + 255) / 256), 256, 0, stream>>>(
      proto, psq, C_CLS * P_PER);

  // Main WMMA kernel: 16 batch-blocks (4 waves x 64 rows) x 63 class tiles.
  proto_dist_kernel<<<dim3(B_ROWS / (4 * R_FRAG * 16), (C_CLS + 15) / 16),
                      128, 0, stream>>>(x, proto, xsq, psq, out);
}